// PhaseVectorMemory_55929064129047
// MI455X (gfx1250) — compile-verified
//
#include <hip/hip_runtime.h>
#include <math.h>

// CDNA5 / gfx1250. Memory-bound chunked-scan + rotation.
// The length-16 intra-chunk scan is computed as D = A*U + C via four chained
// V_WMMA_F32_16X16X4_F32 (K accumulation), carry injected via C, extracted
// from D row 15 with a wave32 shuffle.

typedef __attribute__((ext_vector_type(2))) float v2f;
typedef __attribute__((ext_vector_type(8))) float v8f;

#define ALPHA_  0.12f
#define BETA_   0.88f
#define OMEGA_  6.0f
#define TWO_PI_ 6.28318530717958647692f
#define B_      4
#define T_      4096
#define D_      2048
#define HALF_   1024
#define SEG_    256            // timesteps per segment (pass-1 granularity)
#define NSEG_   (T_ / SEG_)    // 16
#define NCH_    (SEG_ / 16)    // 16 WMMA chunks per segment

// ---------------------------------------------------------------- pass 0 ----
// Phase table: theta(t) = (OMEGA*log1p(t+1+step_idx)) mod 2pi; cos/sin per t.
__global__ void k_theta(float* __restrict__ cs, float* __restrict__ sn,
                        const int* __restrict__ step_idx) {
  int t = blockIdx.x * blockDim.x + threadIdx.x;
  if (t >= T_) return;
  float tg = (float)(t + 1 + step_idx[0]);
  float th = fmodf(OMEGA_ * log1pf(tg), TWO_PI_);
  cs[t] = cosf(th);
  sn[t] = sinf(th);
}

// ---------------------------------------------------------------- pass 1 ----
// Per-(b,d,segment) scan contribution with zero initial state.
// Consecutive threads -> consecutive d: fully coalesced 256-step stream.
__global__ void k_seg_partial(const float* __restrict__ x,
                              const float* __restrict__ input_scale,
                              float* __restrict__ seg_carry) {
  int id  = blockIdx.x * blockDim.x + threadIdx.x;   // B_*D_*NSEG_ threads
  int d   = id % D_;
  int r   = id / D_;
  int b   = r % B_;
  int seg = r / B_;
  float k = ALPHA_ * input_scale[0];
  const float* xp = x + ((size_t)(b * T_ + seg * SEG_)) * D_ + d;
  float m = 0.0f;
  #pragma unroll 4
  for (int i = 0; i < SEG_; ++i)
    m = fmaf(BETA_, m, k * xp[(size_t)i * D_]);
  seg_carry[((size_t)seg * B_ + b) * D_ + d] = m;
}

// ---------------------------------------------------------------- pass 2 ----
// Sequential combine of the 16 segment carries per column (factor BETA^256),
// seeded from mem_state. Emits the carry-in state for every segment.
__global__ void k_carry(const float* __restrict__ seg_carry,
                        const float* __restrict__ mem_state,
                        float* __restrict__ carry_in) {
  int id = blockIdx.x * blockDim.x + threadIdx.x;    // B_*D_ threads
  int d  = id % D_;
  int b  = id / D_;
  float f = powf(BETA_, (float)SEG_);                // ~6e-15 but exact path
  float m = mem_state[d];
  for (int seg = 0; seg < NSEG_; ++seg) {
    size_t o = ((size_t)seg * B_ + b) * D_ + d;
    carry_in[o] = m;
    m = fmaf(f, m, seg_carry[o]);
  }
}

// ---------------------------------------------------------------- pass 3 ----
// One 16-timestep chunk of the scan for one 16-column tile:
//   D = A(16x16 lower-tri BETA powers) * U(16x16) + outer(p, m_in)
// realized as 4 chained v_wmma_f32_16x16x4_f32 (K accumulation in C).
// A lane layout (16x4 f32): lane l holds M=l&15, K = {2*(l>>4), +1}.
// B lane layout (4x16 f32): lane l holds N=l&15, K = {2*(l>>4), +1}.
// C/D layout: lane l holds column N=l&15, rows M = v + 8*(l>>4), v=0..7.
__device__ __forceinline__ v8f scan_tile_wmma(const float* __restrict__ xcol,
                                              float k, const v2f* a,
                                              const float* pw,
                                              float& m_carry, int n, int h) {
  v8f acc;
  #pragma unroll
  for (int v = 0; v < 8; ++v) acc[v] = pw[v] * m_carry;  // carry via C
  #pragma unroll
  for (int q = 0; q < 4; ++q) {
    int kb = 4 * q + 2 * h;
    v2f bb;
    bb.x = k * xcol[(size_t)kb * D_];
    bb.y = k * xcol[(size_t)(kb + 1) * D_];
    acc = __builtin_amdgcn_wmma_f32_16x16x4_f32(
        /*neg_a=*/false, a[q], /*neg_b=*/false, bb,
        /*c_mod=*/(short)0, acc, /*reuse_a=*/false, /*reuse_b=*/false);
  }
  // carry-out = row M=15 -> lane 16+n, VGPR 7; broadcast to both lane halves.
  m_carry = __shfl(acc[7], 16 + n, 32);
  return acc;
}

__global__ void
__launch_bounds__(32)
k_scan_out(const float* __restrict__ x, const float* __restrict__ gate,
           const float* __restrict__ input_scale,
           const float* __restrict__ carry_in,
           const float* __restrict__ cs, const float* __restrict__ sn,
           float* __restrict__ out) {
  const int lane = threadIdx.x;      // one wave32 per block; EXEC all ones
  const int n = lane & 15;
  const int h = lane >> 4;

  int blk = blockIdx.x;              // B_*(HALF_/16)*NSEG_ blocks
  int dt  = blk % (HALF_ / 16);
  int r   = blk / (HALF_ / 16);
  int b   = r % B_;
  int seg = r / B_;

  const float k = ALPHA_ * input_scale[0];
  const float g = 1.0f / (1.0f + expf(-gate[0]));

  // A-operand constants: A[M][kk] = BETA^(M-kk) for kk<=M else 0.
  const int M = n;
  v2f a[4];
  #pragma unroll
  for (int q = 0; q < 4; ++q) {
    int k0 = 4 * q + 2 * h;
    a[q].x = (k0 <= M)     ? powf(BETA_, (float)(M - k0))     : 0.0f;
    a[q].y = (k0 + 1 <= M) ? powf(BETA_, (float)(M - k0 - 1)) : 0.0f;
  }
  // Carry injection weights: C[row i] = BETA^(i+1) * m_in, i = v + 8h.
  float pw[8];
  #pragma unroll
  for (int v = 0; v < 8; ++v) pw[v] = powf(BETA_, (float)(v + 8 * h + 1));

  const int d_lo = dt * 16 + n;
  const int d_hi = d_lo + HALF_;
  const size_t co = ((size_t)seg * B_ + b) * D_;
  float m_lo = carry_in[co + d_lo];
  float m_hi = carry_in[co + d_hi];

  for (int ch = 0; ch < NCH_; ++ch) {
    const int t0 = seg * SEG_ + ch * 16;
    const float* xbase = x + ((size_t)b * T_ + t0) * D_;

    v8f acc_lo = scan_tile_wmma(xbase + d_lo, k, a, pw, m_lo, n, h);
    v8f acc_hi = scan_tile_wmma(xbase + d_hi, k, a, pw, m_hi, n, h);

    // Rotation + residual; D-layout row t = t0 + v + 8h, column n.
    #pragma unroll
    for (int v = 0; v < 8; ++v) {
      int t = t0 + v + 8 * h;
      float ct = cs[t], st = sn[t];
      size_t o = ((size_t)b * T_ + t) * D_;
      float xlo = x[o + d_lo];       // L0/L2 hit: just streamed by WMMA loads
      float xhi = x[o + d_hi];
      float ma = acc_lo[v], mb = acc_hi[v];
      out[o + d_lo] = fmaf(g, fmaf(ma, ct, -mb * st), xlo);
      out[o + d_hi] = fmaf(g, fmaf(ma, st,  mb * ct), xhi);
    }
  }
}

// ----------------------------------------------------------------------------
extern "C" void kernel_launch(void* const* d_in, const int* in_sizes, int n_in,
                              void* d_out, int out_size, void* d_ws,
                              size_t ws_size, hipStream_t stream) {
  const float* x           = (const float*)d_in[0];
  const float* gate        = (const float*)d_in[1];
  const float* input_scale = (const float*)d_in[2];
  const float* mem_state   = (const float*)d_in[3];
  const int*   step_idx    = (const int*)d_in[4];
  float* out = (float*)d_out;

  // Workspace layout (floats): cs[T], sn[T], seg_carry[NSEG*B*D], carry_in[..]
  float* ws        = (float*)d_ws;
  float* cs        = ws;
  float* sn        = ws + T_;
  float* seg_carry = ws + 2 * T_;
  float* carry_in  = seg_carry + (size_t)NSEG_ * B_ * D_;

  k_theta<<<(T_ + 255) / 256, 256, 0, stream>>>(cs, sn, step_idx);
  k_seg_partial<<<(B_ * D_ * NSEG_) / 256, 256, 0, stream>>>(x, input_scale,
                                                             seg_carry);
  k_carry<<<(B_ * D_) / 256, 256, 0, stream>>>(seg_carry, mem_state, carry_in);
  k_scan_out<<<B_ * (HALF_ / 16) * NSEG_, 32, 0, stream>>>(
      x, gate, input_scale, carry_in, cs, sn, out);
}